// AttentionLayer_52012053955019
// MI455X (gfx1250) — compile-verified
//
#include <hip/hip_runtime.h>
#include <hip/hip_bf16.h>
#include <math.h>

// ---------------------------------------------------------------------------
// GATConv forward for MI455X (gfx1250, wave32).
//   xw = x @ W           -> fp32 WMMA (v_wmma_f32_16x16x4_f32)
//   segment softmax over dst + scatter-add aggregation -> L2 atomics
// ---------------------------------------------------------------------------

typedef float v2f __attribute__((ext_vector_type(2)));
typedef float v8f __attribute__((ext_vector_type(8)));

#define IN_C   128
#define OUTHC  256      // HEADS * OUT_C
#define HEADS  8
#define OUT_C  32
#define NEG_SLOPE 0.2f

// ---- monotonic float<->uint encoding for atomicMax on floats ---------------
__device__ __forceinline__ unsigned enc_f32(float f) {
    unsigned u = __float_as_uint(f);
    return (f >= 0.0f) ? (u | 0x80000000u) : ~u;
}
__device__ __forceinline__ float dec_f32(unsigned u) {
    return (u & 0x80000000u) ? __uint_as_float(u & 0x7FFFFFFFu)
                             : __uint_as_float(~u);
}
#define ENC_NEG_INF 0x007FFFFFu   // enc_f32(-inf)

// ---------------------------------------------------------------------------
// 0) init: out = bias (broadcast), amax = enc(-inf), denom = 0
// ---------------------------------------------------------------------------
__global__ __launch_bounds__(256) void gat_init(float* __restrict__ out,
                                                const float* __restrict__ bias,
                                                unsigned* __restrict__ amax,
                                                float* __restrict__ denom,
                                                int n) {
    long long idx = (long long)blockIdx.x * blockDim.x + threadIdx.x;
    long long total = (long long)n * OUTHC;
    if (idx < total) {
        out[idx] = bias[idx % OUTHC];
    }
    long long nh = (long long)n * HEADS;
    if (idx < nh) {
        amax[idx]  = ENC_NEG_INF;
        denom[idx] = 0.0f;
    }
}

// ---------------------------------------------------------------------------
// 1) GEMM: xw[n,256] = x[n,128] @ W[128,256] using V_WMMA_F32_16X16X4_F32.
//    One wave computes a 16x64 strip: 1 M-tile x 4 N-tiles.
//      - A fragment loaded once per k-step, reused by 4 WMMAs (4x less A BW)
//      - 4 independent accumulators -> no WMMA->WMMA RAW chain
//    A fragment (16x4 fp32, 2 VGPRs/lane): lanes 0-15 -> K={k,k+1},
//      lanes 16-31 -> K={k+2,k+3}, M = lane&15.
//    B fragment (4x16 fp32): same K split, N = lane&15.
//    C/D (16x16 fp32, 8 VGPRs): N = lane&15, M = v + 8*(lane>>4).
// ---------------------------------------------------------------------------
__global__ __launch_bounds__(256) void gat_gemm_wmma(const float* __restrict__ x,
                                                     const float* __restrict__ W,
                                                     float* __restrict__ xw,
                                                     int n, int mtiles) {
    const int lane  = threadIdx.x & 31;
    const int waveG = blockIdx.x * 8 + (threadIdx.x >> 5);
    const int mt  = waveG >> 2;         // M-tile index
    const int ng  = waveG & 3;          // N-tile group (4 tiles of 16 -> 64 cols)
    if (mt >= mtiles) return;           // wave-uniform: EXEC stays all-ones

    const int m     = lane & 15;
    const int row   = mt * 16 + m;
    const int rowC  = (row < n) ? row : (n - 1);    // clamp tail reads
    const int col0  = ng * 64 + m;                  // base column for this lane
    const int koff  = (lane < 16) ? 0 : 2;

    const float* xrow = x + (long long)rowC * IN_C;

    v8f acc0 = {0.f,0.f,0.f,0.f,0.f,0.f,0.f,0.f};
    v8f acc1 = acc0, acc2 = acc0, acc3 = acc0;

    #pragma unroll 2
    for (int k = 0; k < IN_C; k += 4) {
        v2f a;
        a.x = xrow[k + koff + 0];
        a.y = xrow[k + koff + 1];
        const float* w0 = W + (long long)(k + koff) * OUTHC + col0;
        const float* w1 = w0 + OUTHC;
        v2f b0; b0.x = w0[ 0]; b0.y = w1[ 0];
        v2f b1; b1.x = w0[16]; b1.y = w1[16];
        v2f b2; b2.x = w0[32]; b2.y = w1[32];
        v2f b3; b3.x = w0[48]; b3.y = w1[48];
        acc0 = __builtin_amdgcn_wmma_f32_16x16x4_f32(false, a, false, b0,
                                                     (short)0, acc0, false, false);
        acc1 = __builtin_amdgcn_wmma_f32_16x16x4_f32(false, a, false, b1,
                                                     (short)0, acc1, false, false);
        acc2 = __builtin_amdgcn_wmma_f32_16x16x4_f32(false, a, false, b2,
                                                     (short)0, acc2, false, false);
        acc3 = __builtin_amdgcn_wmma_f32_16x16x4_f32(false, a, false, b3,
                                                     (short)0, acc3, false, false);
    }

    // store: lane holds column N, rows rBase..rBase+7
    const int rBase = mt * 16 + ((lane >> 4) << 3);   // +0 or +8
    float* o0 = xw + (long long)rBase * OUTHC + col0; // tile 0 column
    if (mt * 16 + 16 <= n) {                          // full tile: no guards
        #pragma unroll
        for (int v = 0; v < 8; ++v) {
            float* orow = o0 + (long long)v * OUTHC;
            orow[ 0] = acc0[v];
            orow[16] = acc1[v];
            orow[32] = acc2[v];
            orow[48] = acc3[v];
        }
    } else {                                          // tail tile: guarded
        #pragma unroll
        for (int v = 0; v < 8; ++v) {
            if (rBase + v < n) {
                float* orow = o0 + (long long)v * OUTHC;
                orow[ 0] = acc0[v];
                orow[16] = acc1[v];
                orow[32] = acc2[v];
                orow[48] = acc3[v];
            }
        }
    }
}

// ---------------------------------------------------------------------------
// 2) per-(node,head) logits: a_src[n,h] = dot(xw[n,h,:], att_src[h,:])
// ---------------------------------------------------------------------------
__global__ __launch_bounds__(256) void gat_logits(const float* __restrict__ xw,
                                                  const float* __restrict__ att_src,
                                                  const float* __restrict__ att_dst,
                                                  float* __restrict__ a_src,
                                                  float* __restrict__ a_dst,
                                                  int n) {
    long long idx = (long long)blockIdx.x * blockDim.x + threadIdx.x;
    if (idx >= (long long)n * HEADS) return;
    int node = (int)(idx >> 3);
    int h    = (int)(idx & 7);
    const float* v  = xw + (long long)node * OUTHC + h * OUT_C;
    const float* as = att_src + h * OUT_C;
    const float* ad = att_dst + h * OUT_C;
    float s = 0.f, d = 0.f;
    #pragma unroll
    for (int c = 0; c < OUT_C; ++c) { s += v[c] * as[c]; d += v[c] * ad[c]; }
    a_src[idx] = s;
    a_dst[idx] = d;
}

// ---------------------------------------------------------------------------
// 3) edge pass 1: raw = leakyrelu(a_src[src]+a_dst[dst]); atomicMax amax[dst]
//    edges e<E come from edge_index; e>=E are self-loops (src=dst=e-E)
// ---------------------------------------------------------------------------
__global__ __launch_bounds__(256) void gat_edge_max(const int* __restrict__ ei,
                                                    const float* __restrict__ a_src,
                                                    const float* __restrict__ a_dst,
                                                    float* __restrict__ ebuf,
                                                    unsigned* __restrict__ amax,
                                                    int E, int Etot) {
    long long idx = (long long)blockIdx.x * blockDim.x + threadIdx.x;
    if (idx >= (long long)Etot * HEADS) return;
    int e = (int)(idx >> 3);
    int h = (int)(idx & 7);
    int src = (e < E) ? ei[e]     : (e - E);
    int dst = (e < E) ? ei[E + e] : (e - E);
    float raw = a_src[(long long)src * HEADS + h] + a_dst[(long long)dst * HEADS + h];
    float lr  = (raw > 0.f) ? raw : NEG_SLOPE * raw;
    ebuf[idx] = lr;
    atomicMax(&amax[(long long)dst * HEADS + h], enc_f32(lr));
}

// ---------------------------------------------------------------------------
// 4) edge pass 2: ex = exp(raw - amax[dst]); atomicAdd denom[dst]
// ---------------------------------------------------------------------------
__global__ __launch_bounds__(256) void gat_edge_exp(const int* __restrict__ ei,
                                                    float* __restrict__ ebuf,
                                                    const unsigned* __restrict__ amax,
                                                    float* __restrict__ denom,
                                                    int E, int Etot) {
    long long idx = (long long)blockIdx.x * blockDim.x + threadIdx.x;
    if (idx >= (long long)Etot * HEADS) return;
    int e = (int)(idx >> 3);
    int h = (int)(idx & 7);
    int dst = (e < E) ? ei[E + e] : (e - E);
    float mx = dec_f32(amax[(long long)dst * HEADS + h]);
    float ex = __expf(ebuf[idx] - mx);
    ebuf[idx] = ex;
    atomicAdd(&denom[(long long)dst * HEADS + h], ex);
}

// ---------------------------------------------------------------------------
// 5) edge pass 3: one 256-thread block per edge; thread t handles channel t.
//    out[dst, t] += xw[src, t] * (ex / (denom[dst,h]+1e-16))
//    xw[src] read is fully coalesced (256 contiguous floats / block) and
//    L2-resident; the atomic adds resolve in L2 atomic units.
// ---------------------------------------------------------------------------
__global__ __launch_bounds__(256) void gat_edge_agg(const int* __restrict__ ei,
                                                    const float* __restrict__ xw,
                                                    const float* __restrict__ ebuf,
                                                    const float* __restrict__ denom,
                                                    float* __restrict__ out,
                                                    int E, int Etot) {
    int e = blockIdx.x;
    if (e >= Etot) return;
    int t = threadIdx.x;                 // 0..255
    int h = t >> 5;                      // head
    int src = (e < E) ? ei[e]     : (e - E);
    int dst = (e < E) ? ei[E + e] : (e - E);
    float ex = ebuf[(long long)e * HEADS + h];
    float dn = denom[(long long)dst * HEADS + h] + 1e-16f;
    float w  = ex / dn;
    float msg = xw[(long long)src * OUTHC + t] * w;
    atomicAdd(&out[(long long)dst * OUTHC + t], msg);
}

// ---------------------------------------------------------------------------
extern "C" void kernel_launch(void* const* d_in, const int* in_sizes, int n_in,
                              void* d_out, int out_size, void* d_ws, size_t ws_size,
                              hipStream_t stream) {
    const float* x       = (const float*)d_in[0];   // [N,128]
    const int*   ei      = (const int*)  d_in[1];   // [2,E]
    const float* W       = (const float*)d_in[2];   // [128,256]
    const float* att_src = (const float*)d_in[3];   // [1,8,32] -> 256
    const float* att_dst = (const float*)d_in[4];   // [1,8,32] -> 256
    const float* bias    = (const float*)d_in[5];   // [256]
    float* out = (float*)d_out;

    const int N    = in_sizes[0] / IN_C;
    const int E    = in_sizes[1] / 2;
    const int Etot = E + N;

    // workspace layout (fp32 elements)
    float*    xw    = (float*)d_ws;                           // N*256
    float*    a_src = xw    + (size_t)N * OUTHC;              // N*8
    float*    a_dst = a_src + (size_t)N * HEADS;              // N*8
    unsigned* amax  = (unsigned*)(a_dst + (size_t)N * HEADS); // N*8
    float*    denom = (float*)(amax + (size_t)N * HEADS);     // N*8
    float*    ebuf  = denom + (size_t)N * HEADS;              // Etot*8

    // 0) init out/amax/denom
    {
        long long total = (long long)N * OUTHC;
        int blocks = (int)((total + 255) / 256);
        gat_init<<<blocks, 256, 0, stream>>>(out, bias, amax, denom, N);
    }
    // 1) WMMA GEMM (one wave per 16x64 strip: mtiles * 4 waves)
    {
        int mtiles = (N + 15) / 16;
        long long waves = (long long)mtiles * 4;
        int blocks = (int)((waves + 7) / 8);        // 8 waves / 256-thread block
        gat_gemm_wmma<<<blocks, 256, 0, stream>>>(x, W, xw, N, mtiles);
    }
    // 2) logits
    {
        long long total = (long long)N * HEADS;
        int blocks = (int)((total + 255) / 256);
        gat_logits<<<blocks, 256, 0, stream>>>(xw, att_src, att_dst, a_src, a_dst, N);
    }
    // 3) segment max
    {
        long long total = (long long)Etot * HEADS;
        int blocks = (int)((total + 255) / 256);
        gat_edge_max<<<blocks, 256, 0, stream>>>(ei, a_src, a_dst, ebuf, amax, E, Etot);
    }
    // 4) exp + segment sum
    {
        long long total = (long long)Etot * HEADS;
        int blocks = (int)((total + 255) / 256);
        gat_edge_exp<<<blocks, 256, 0, stream>>>(ei, ebuf, amax, denom, E, Etot);
    }
    // 5) weighted scatter-add
    {
        gat_edge_agg<<<Etot, 256, 0, stream>>>(ei, xw, ebuf, denom, out, E, Etot);
    }
}